// RQSBijector_79104707658012
// MI455X (gfx1250) — compile-verified
//
#include <hip/hip_runtime.h>
#include <math.h>

typedef float v4f __attribute__((ext_vector_type(4)));

#define RQS_RANGE_MIN  (-5.0f)
#define RQS_RANGE_MAX  ( 5.0f)
#define RQS_MIN_BIN    (0.0001f)
#define RQS_MIN_SLOPE  (0.0001f)
#define RQS_K          8
#define LN2F           0.69314718055994530942f

// Per-element spline evaluation. Bin table lives in LDS:
//  bin0[b] = {x_k, 1/width, y_k, height}
//  bin1[b] = {s = h/w, slope_k, slope_k1, 0}
__device__ __forceinline__ void rqs_eval(
    float xc,
    float k1, float k2, float k3, float k4, float k5, float k6, float k7,
    const v4f* __restrict__ s_bin0, const v4f* __restrict__ s_bin1,
    float slope_l, float slope_r, float log_sl, float log_sr,
    float& y_out, float& ld_out)
{
    // searchsorted(x_pos[1:-1], x, 'right') == #(knot <= x)
    int b = (xc >= k1) + (xc >= k2) + (xc >= k3) + (xc >= k4)
          + (xc >= k5) + (xc >= k6) + (xc >= k7);

    v4f c0 = s_bin0[b];   // ds_load_b128, conflict-free (bin*4 banks)
    v4f c1 = s_bin1[b];

    float xi = (xc - c0[0]) * c0[1];
    xi = fminf(fmaxf(xi, 0.0f), 1.0f);
    float om = 1.0f - xi;
    float xm = xi * om;

    float s   = c1[0];
    float sk  = c1[1];
    float sk1 = c1[2];

    float num = fmaf(s * xi, xi, sk * xm);                 // s*xi^2 + sk*xi*(1-xi)
    float den = fmaf(sk1 + sk - 2.0f * s, xm, s);          // s + (sk1+sk-2s)*xi*(1-xi)
    float r   = __builtin_amdgcn_rcpf(den);                // v_rcp_f32, reused twice

    float ys  = fmaf(c0[3] * num, r, c0[2]);               // y_k + h*num/den

    // s^2 * (sk1*xi^2 + 2s*xi*(1-xi) + sk*(1-xi)^2)
    float nd  = fmaf(sk1 * xi, xi, fmaf(2.0f * s, xm, sk * om * om)) * (s * s);
    float ld  = __builtin_amdgcn_logf(nd * (r * r)) * LN2F; // v_log_f32 (log2) -> ln

    bool below = xc < RQS_RANGE_MIN;
    bool above = xc > RQS_RANGE_MAX;
    float yl = fmaf(xc - RQS_RANGE_MIN, slope_l, RQS_RANGE_MIN);
    float yr = fmaf(xc - RQS_RANGE_MAX, slope_r, RQS_RANGE_MAX);

    ys = above ? yr     : ys;
    ys = below ? yl     : ys;
    ld = above ? log_sr : ld;
    ld = below ? log_sl : ld;

    y_out  = ys;
    ld_out = ld;
}

__global__ __launch_bounds__(256) void rqs_bijector_kernel(
    const float* __restrict__ x, const float* __restrict__ params,
    float* __restrict__ out_y, float* __restrict__ out_ld, long long n)
{
    __shared__ float s_knots[RQS_K];       // x_pos[0..7] (search uses 1..7)
    __shared__ v4f   s_bin0[RQS_K];
    __shared__ v4f   s_bin1[RQS_K];
    __shared__ float s_misc[4];            // slope0, slope8, log(slope0), log(slope8)

    if (threadIdx.x == 0) {
        float p[3 * RQS_K + 1];
        #pragma unroll
        for (int i = 0; i < 3 * RQS_K + 1; ++i) p[i] = params[i];

        const float total = RQS_RANGE_MAX - RQS_RANGE_MIN;
        float w[RQS_K], h[RQS_K], sl[RQS_K + 1];

        {   // widths = softmax(p[0:8]) * (10 - 8*minbin) + minbin   (stable softmax)
            float m = p[0];
            for (int i = 1; i < RQS_K; ++i) m = fmaxf(m, p[i]);
            float sum = 0.0f;
            for (int i = 0; i < RQS_K; ++i) { w[i] = expf(p[i] - m); sum += w[i]; }
            float scale = (total - RQS_K * RQS_MIN_BIN) / sum;
            for (int i = 0; i < RQS_K; ++i) w[i] = w[i] * scale + RQS_MIN_BIN;
        }
        {   // heights = softmax(p[8:16]) * ... + minbin
            float m = p[RQS_K];
            for (int i = 1; i < RQS_K; ++i) m = fmaxf(m, p[RQS_K + i]);
            float sum = 0.0f;
            for (int i = 0; i < RQS_K; ++i) { h[i] = expf(p[RQS_K + i] - m); sum += h[i]; }
            float scale = (total - RQS_K * RQS_MIN_BIN) / sum;
            for (int i = 0; i < RQS_K; ++i) h[i] = h[i] * scale + RQS_MIN_BIN;
        }
        {   // slopes = softplus(p[16:25] + off) + min_slope  (jax stable softplus)
            float off = logf(expf(1.0f - RQS_MIN_SLOPE) - 1.0f);
            for (int i = 0; i <= RQS_K; ++i) {
                float z  = p[2 * RQS_K + i] + off;
                float sp = fmaxf(z, 0.0f) + log1pf(expf(-fabsf(z)));
                sl[i] = sp + RQS_MIN_SLOPE;
            }
        }

        // x_pos/y_pos = concat([0], cumsum) + RANGE_MIN  (match jnp rounding order)
        float xpos[RQS_K + 1], ypos[RQS_K + 1];
        xpos[0] = RQS_RANGE_MIN; ypos[0] = RQS_RANGE_MIN;
        float cx = 0.0f, cy = 0.0f;
        for (int i = 0; i < RQS_K; ++i) {
            cx += w[i]; xpos[i + 1] = cx + RQS_RANGE_MIN;
            cy += h[i]; ypos[i + 1] = cy + RQS_RANGE_MIN;
        }

        for (int b = 0; b < RQS_K; ++b) {
            float xk = xpos[b], wk = xpos[b + 1] - xk;
            float yk = ypos[b], hk = ypos[b + 1] - yk;
            v4f c0; c0[0] = xk;      c0[1] = 1.0f / wk; c0[2] = yk;        c0[3] = hk;
            v4f c1; c1[0] = hk / wk; c1[1] = sl[b];     c1[2] = sl[b + 1]; c1[3] = 0.0f;
            s_bin0[b]  = c0;
            s_bin1[b]  = c1;
            s_knots[b] = xpos[b];
        }
        s_misc[0] = sl[0];
        s_misc[1] = sl[RQS_K];
        s_misc[2] = logf(sl[0]);
        s_misc[3] = logf(sl[RQS_K]);
    }
    __syncthreads();

    // Interior knots + edge constants into registers (read once from LDS)
    const float k1 = s_knots[1], k2 = s_knots[2], k3 = s_knots[3], k4 = s_knots[4],
                k5 = s_knots[5], k6 = s_knots[6], k7 = s_knots[7];
    const float slope_l = s_misc[0], slope_r = s_misc[1];
    const float log_sl  = s_misc[2], log_sr  = s_misc[3];

    const long long nvec   = n >> 2;
    const long long stride = (long long)gridDim.x * blockDim.x;
    long long tid = (long long)blockIdx.x * blockDim.x + threadIdx.x;

    const v4f* __restrict__ x4 = (const v4f*)x;
    v4f* __restrict__ y4 = (v4f*)out_y;
    v4f* __restrict__ l4 = (v4f*)out_ld;

    for (long long i = tid; i < nvec; i += stride) {
        v4f xv = __builtin_nontemporal_load(x4 + i);   // global_load_b128 th:NT
        v4f yo, lo;
        #pragma unroll
        for (int c = 0; c < 4; ++c) {
            float ys, ld;
            rqs_eval(xv[c], k1, k2, k3, k4, k5, k6, k7,
                     s_bin0, s_bin1, slope_l, slope_r, log_sl, log_sr, ys, ld);
            yo[c] = ys; lo[c] = ld;
        }
        __builtin_nontemporal_store(yo, y4 + i);       // global_store_b128 th:NT
        __builtin_nontemporal_store(lo, l4 + i);
    }

    // scalar tail (N here is divisible by 4; kept for generality)
    for (long long i = (nvec << 2) + tid; i < n; i += stride) {
        float ys, ld;
        rqs_eval(x[i], k1, k2, k3, k4, k5, k6, k7,
                 s_bin0, s_bin1, slope_l, slope_r, log_sl, log_sr, ys, ld);
        __builtin_nontemporal_store(ys, out_y + i);
        __builtin_nontemporal_store(ld, out_ld + i);
    }
}

extern "C" void kernel_launch(void* const* d_in, const int* in_sizes, int n_in,
                              void* d_out, int out_size, void* d_ws, size_t ws_size,
                              hipStream_t stream) {
    (void)d_ws; (void)ws_size; (void)n_in;
    const float* x      = (const float*)d_in[0];
    const float* params = (const float*)d_in[1];
    long long n = (long long)in_sizes[0];

    float* out_y  = (float*)d_out;          // outputs concatenated: [y(N), logdet(N)]
    float* out_ld = (float*)d_out + n;

    const int threads = 256;                      // 8 wave32 per block
    long long nvec = n >> 2;
    long long want = (nvec + threads - 1) / threads;
    int blocks = (int)(want < 8192 ? (want > 0 ? want : 1) : 8192);

    rqs_bijector_kernel<<<blocks, threads, 0, stream>>>(x, params, out_y, out_ld, n);
}